// KoopmanQNetwork_75952201662835
// MI455X (gfx1250) — compile-verified
//
#include <hip/hip_runtime.h>

typedef __attribute__((ext_vector_type(2))) float v2f;
typedef __attribute__((ext_vector_type(8))) float v8f;

#define KD_PHI 128
#define KD_PSI 8
#define KSDIM  17
#define KADIM  7
#define MSIZE  (KD_PHI * KD_PSI)      // 1024 floats
#define CPAD_K 20                     // K padded 17 -> 20 (5 wmma k-steps of 4)
#define CPAD_N 16                     // N padded 7 -> 16
#define CPAD_SIZE (CPAD_K * CPAD_N)   // 320 floats
// ws float layout: [0, MSIZE) = M[j][z] ; [MSIZE, MSIZE+CPAD_SIZE) = Cpad[k][n]

__global__ void koop_zero_ws(float* __restrict__ ws) {
    int t = blockIdx.x * blockDim.x + threadIdx.x;
    if (t < MSIZE + CPAD_SIZE) ws[t] = 0.0f;
}

// M[j,z] = sum_i wlin[i] * K[i][j][z].  Block b reduces i in [4b,4b+4) over all
// 1024 outputs (coalesced 4KB row reads), atomicAdd into ws so the 512KB scan
// of K is spread across 32 workgroups instead of serializing on one WGP.
__global__ __launch_bounds__(256) void koop_fold_M(const float* __restrict__ K,
                                                   const float* __restrict__ wlin,
                                                   float* __restrict__ wsM) {
    const int i0 = blockIdx.x * 4;
    const int t  = threadIdx.x;
#pragma unroll
    for (int c = 0; c < 4; ++c) {
        const int o = t + c * 256;
        float acc = 0.0f;
#pragma unroll
        for (int ii = 0; ii < 4; ++ii) {
            const int i = i0 + ii;
            acc = fmaf(wlin[i], K[i * MSIZE + o], acc);
        }
        atomicAdd(&wsM[o], acc);
    }
}

// Cpad[s][a] = sum_j W_phi[j,s] * ( sum_z M[j,z] * W_psi[z,a] ); padding stays 0.
__global__ __launch_bounds__(128) void koop_fold_C(const float* __restrict__ Wphi,
                                                   const float* __restrict__ Wpsi,
                                                   const float* __restrict__ wsM,
                                                   float* __restrict__ Cpad) {
    const int t = threadIdx.x;
    if (t >= KSDIM * KADIM) return;
    const int s = t / KADIM, a = t % KADIM;
    float c = 0.0f;
    for (int j = 0; j < KD_PHI; ++j) {
        float inner = 0.0f;
#pragma unroll
        for (int z = 0; z < KD_PSI; ++z)
            inner = fmaf(wsM[j * KD_PSI + z], Wpsi[z * KADIM + a], inner);
        c = fmaf(Wphi[j * KSDIM + s], inner, c);
    }
    Cpad[s * CPAD_N + a] = c;
}

// Streaming phase: per wave, T(16x16) = state_tile(16x20) @ Cpad(20x16) via five
// V_WMMA_F32_16X16X4_F32, then out[m] = sum_{a<7} T[m][a]*action[m][a] through a
// wave-private LDS tile.  All A loads are unconditional (row index clamped into
// range, out-of-K slots zeroed by value-select) so the hot loop has no exec-mask
// branches: 10 batched global_load_b32 -> one loadcnt wait -> 5 chained WMMAs.
__global__ __launch_bounds__(256) void koop_main(const float* __restrict__ state,
                                                 const float* __restrict__ action,
                                                 const float* __restrict__ Cpad,
                                                 float* __restrict__ out,
                                                 int nB, int ntiles) {
    __shared__ float tileD[8][256];            // 8 waves/block * 16x16 f32 tile
    const int lane = threadIdx.x & 31;
    const int wave = threadIdx.x >> 5;
    const int half = lane >> 4;                // A/B: lanes 16-31 hold K+2/K+3
    const int l16  = lane & 15;

    // Constant B fragments (Cpad) held in VGPRs for the whole kernel.
    v2f bfrag[5];
#pragma unroll
    for (int ks = 0; ks < 5; ++ks) {
        const int k = ks * 4 + 2 * half;
        v2f b;
        b.x = Cpad[k * CPAD_N + l16];
        b.y = Cpad[(k + 1) * CPAD_N + l16];
        bfrag[ks] = b;
    }

    const int waveGid    = (blockIdx.x * blockDim.x + threadIdx.x) >> 5;
    const int wavesTotal = (gridDim.x * blockDim.x) >> 5;

    for (int tile = waveGid; tile < ntiles; tile += wavesTotal) {
        const int m0 = tile * 16;
        const int m  = m0 + l16;               // row owned by this lane-half
        // Clamp to a valid row instead of predicating loads (no-op when 16|B).
        const int mc = (m < nB) ? m : (nB - 1);
        const float* srow = state + (size_t)mc * KSDIM;

        // Gather all A fragments first: unconditional loads, value-zero the
        // padded K slots (only the last k-step has a runtime, half-dependent
        // condition -> v_cndmask, never an exec branch).
        v2f afrag[5];
#pragma unroll
        for (int ks = 0; ks < 5; ++ks) {
            const int kx = ks * 4 + 2 * half;
            const int ky = kx + 1;
            const float vx = srow[(kx <= KSDIM - 1) ? kx : (KSDIM - 1)];
            const float vy = srow[(ky <= KSDIM - 1) ? ky : (KSDIM - 1)];
            v2f a;
            a.x = (kx <= KSDIM - 1) ? vx : 0.0f;
            a.y = (ky <= KSDIM - 1) ? vy : 0.0f;
            afrag[ks] = a;
        }

        v8f c = {};
#pragma unroll
        for (int ks = 0; ks < 5; ++ks) {
            // (neg_a, A, neg_b, B, c_mod, C, reuse_a, reuse_b)
            c = __builtin_amdgcn_wmma_f32_16x16x4_f32(false, afrag[ks], false,
                                                      bfrag[ks], (short)0, c,
                                                      false, false);
        }

        // D layout: VGPR r, lanes 0-15 -> row r, lanes 16-31 -> row r+8.
#pragma unroll
        for (int r = 0; r < 8; ++r)
            tileD[wave][(r + 8 * half) * 16 + l16] = c[r];
        // Wave-private LDS; per-wave DS ops are in-order, just fence compiler +
        // wait out the stores before cross-lane reads.
        asm volatile("s_wait_dscnt 0x0" ::: "memory");

        if (lane < 16 && m < nB) {
            const float* arow = action + (size_t)m * KADIM;
            float acc = 0.0f;
#pragma unroll
            for (int a2 = 0; a2 < KADIM; ++a2)
                acc = fmaf(tileD[wave][l16 * 16 + a2], arow[a2], acc);
            out[m] = acc;                       // lanes 0-15: coalesced 64B store
        }
        asm volatile("s_wait_dscnt 0x0" ::: "memory");
    }
}

extern "C" void kernel_launch(void* const* d_in, const int* in_sizes, int n_in,
                              void* d_out, int out_size, void* d_ws, size_t ws_size,
                              hipStream_t stream) {
    const float* state  = (const float*)d_in[0];
    const float* action = (const float*)d_in[1];
    const float* Wphi   = (const float*)d_in[2];
    const float* Wpsi   = (const float*)d_in[3];
    const float* K      = (const float*)d_in[4];
    const float* wlin   = (const float*)d_in[5];
    float* out  = (float*)d_out;
    float* ws   = (float*)d_ws;
    float* wsM  = ws;
    float* Cpad = ws + MSIZE;

    const int B      = in_sizes[0] / KSDIM;
    const int ntiles = (B + 15) / 16;

    koop_zero_ws<<<(MSIZE + CPAD_SIZE + 255) / 256, 256, 0, stream>>>(ws);
    koop_fold_M<<<KD_PHI / 4, 256, 0, stream>>>(K, wlin, wsM);
    koop_fold_C<<<1, 128, 0, stream>>>(Wphi, Wpsi, wsM, Cpad);

    int blocks = (ntiles + 7) / 8;             // 8 waves per 256-thread block
    if (blocks > 4096) blocks = 4096;          // grid-stride covers remainder
    koop_main<<<blocks, 256, 0, stream>>>(state, action, Cpad, out, B, ntiles);
}